// PPO_GNN_Dual_LSTM_42932493091171
// MI455X (gfx1250) — compile-verified
//
#include <hip/hip_runtime.h>
#include <math.h>

// ---- problem constants (from reference) ----
#define SS   64                  // sequence length / "batch" S
#define NNODE 400                // nodes N
#define EE   8000                // edges E
#define NED  (EE + NNODE)        // edges incl. self loops = 8400
#define HH   4                   // heads
#define CCH  32                  // channels per head
#define EMB  128                 // H*C
#define G4   (4*EMB)             // 512 LSTM gate width
#define SN   (SS*NNODE)          // 25600 rows
#define ACT  (SN*EMB)            // 3276800 activation elems
#define HC   (NNODE*EMB)         // 51200 hidden state elems
#define NEG_SLOPE 0.2f

typedef __attribute__((ext_vector_type(2))) float v2f;
typedef __attribute__((ext_vector_type(8))) float v8f;

// ---------------- helpers ----------------
__device__ __forceinline__ unsigned int ordf(float f) {
  unsigned int u = __float_as_uint(f);
  return (u & 0x80000000u) ? ~u : (u | 0x80000000u);
}
__device__ __forceinline__ float unordf(unsigned int o) {
  unsigned int u = (o & 0x80000000u) ? (o & 0x7FFFFFFFu) : ~o;
  return __uint_as_float(u);
}
__device__ __forceinline__ float sigf(float x) { return 1.0f / (1.0f + expf(-x)); }

// ---------------- WMMA fp32 NT GEMM, 4-way N register blocking ----------------
// D[M x N] = A[M x K] * Bt[N x K]^T  (+ A2[M x K2] * Bt2[N x K2]^T) + bias, optional relu.
// One wave32 computes a 16x64 output strip (4 consecutive 16x16 tiles) so the A
// fragment is reused 4x per K-step: 1 A load + 4 B loads + 4 v_wmma.
// Requires: M % 16 == 0, N % 64 == 0, K % 4 == 0 (true for every GEMM here).
// ISA VGPR layouts (CDNA5 ISA):
//   A 16x4: lane<16 -> {A[l][k0],A[l][k0+1]}, lane>=16 -> {A[l-16][k0+2],A[l-16][k0+3]}
//   B 4x16: lane<16 -> {B[k0][l],B[k0+1][l]},  lane>=16 -> {B[k0+2][l-16],B[k0+3][l-16]}
//   D 16x16: VGPR v -> row v (lanes 0-15), row v+8 (lanes 16-31)
__global__ void gemm_nt_wmma(const float* __restrict__ A, const float* __restrict__ Bt,
                             int K, int lda, int ldb,
                             const float* __restrict__ A2, const float* __restrict__ Bt2,
                             int K2, int lda2, int ldb2,
                             const float* __restrict__ bias,
                             float* __restrict__ D, int ldd, int M, int N, int do_relu)
{
  const int lane = threadIdx.x & 31;
  const int wave = (blockIdx.x * blockDim.x + threadIdx.x) >> 5;
  const int grpN = N >> 6;                      // groups of 4 N-tiles
  const int grps = (M >> 4) * grpN;
  if (wave >= grps) return;                     // wave-uniform: EXEC stays all-1s for WMMA
  const int tm   = wave / grpN;
  const int tg   = wave - tm * grpN;
  const int half = lane >> 4;
  const int l15  = lane & 15;
  const int ar   = tm * 16 + l15;               // A row held by this lane
  const int bc0  = tg * 64 + l15;               // first B column held by this lane

  v8f acc[4];
#pragma unroll
  for (int i = 0; i < 4; ++i) acc[i] = (v8f){0.f,0.f,0.f,0.f,0.f,0.f,0.f,0.f};

  {
    const float* ap = A + (size_t)ar * lda;
    const float* bp0 = Bt + (size_t)(bc0 +  0) * ldb;
    const float* bp1 = Bt + (size_t)(bc0 + 16) * ldb;
    const float* bp2 = Bt + (size_t)(bc0 + 32) * ldb;
    const float* bp3 = Bt + (size_t)(bc0 + 48) * ldb;
    for (int k0 = 0; k0 < K; k0 += 4) {
      const int kk = k0 + half * 2;
      v2f av = *(const v2f*)(ap + kk);
      v2f b0 = *(const v2f*)(bp0 + kk);
      v2f b1 = *(const v2f*)(bp1 + kk);
      v2f b2 = *(const v2f*)(bp2 + kk);
      v2f b3 = *(const v2f*)(bp3 + kk);
      acc[0] = __builtin_amdgcn_wmma_f32_16x16x4_f32(false, av, false, b0, (short)0, acc[0], false, false);
      acc[1] = __builtin_amdgcn_wmma_f32_16x16x4_f32(false, av, false, b1, (short)0, acc[1], false, false);
      acc[2] = __builtin_amdgcn_wmma_f32_16x16x4_f32(false, av, false, b2, (short)0, acc[2], false, false);
      acc[3] = __builtin_amdgcn_wmma_f32_16x16x4_f32(false, av, false, b3, (short)0, acc[3], false, false);
    }
  }
  if (A2) {                                     // fused second accumulate (LSTM: x@Wih^T + h@Whh^T)
    const float* ap = A2 + (size_t)ar * lda2;
    const float* bp0 = Bt2 + (size_t)(bc0 +  0) * ldb2;
    const float* bp1 = Bt2 + (size_t)(bc0 + 16) * ldb2;
    const float* bp2 = Bt2 + (size_t)(bc0 + 32) * ldb2;
    const float* bp3 = Bt2 + (size_t)(bc0 + 48) * ldb2;
    for (int k0 = 0; k0 < K2; k0 += 4) {
      const int kk = k0 + half * 2;
      v2f av = *(const v2f*)(ap + kk);
      v2f b0 = *(const v2f*)(bp0 + kk);
      v2f b1 = *(const v2f*)(bp1 + kk);
      v2f b2 = *(const v2f*)(bp2 + kk);
      v2f b3 = *(const v2f*)(bp3 + kk);
      acc[0] = __builtin_amdgcn_wmma_f32_16x16x4_f32(false, av, false, b0, (short)0, acc[0], false, false);
      acc[1] = __builtin_amdgcn_wmma_f32_16x16x4_f32(false, av, false, b1, (short)0, acc[1], false, false);
      acc[2] = __builtin_amdgcn_wmma_f32_16x16x4_f32(false, av, false, b2, (short)0, acc[2], false, false);
      acc[3] = __builtin_amdgcn_wmma_f32_16x16x4_f32(false, av, false, b3, (short)0, acc[3], false, false);
    }
  }
#pragma unroll
  for (int i = 0; i < 4; ++i) {
    const int bc = bc0 + i * 16;
    const float bval = bias ? bias[bc] : 0.f;
#pragma unroll
    for (int v = 0; v < 8; ++v) {
      float val = acc[i][v] + bval;
      if (do_relu) val = fmaxf(val, 0.f);
      D[(size_t)(tm * 16 + v + half * 8) * ldd + bc] = val;
    }
  }
}

// ---------------- small utility kernels ----------------
__global__ void build_edges(const int* __restrict__ ei, int* __restrict__ src, int* __restrict__ dst) {
  int e = blockIdx.x * blockDim.x + threadIdx.x;
  if (e >= NED) return;
  if (e < EE) { src[e] = ei[e]; dst[e] = ei[EE + e]; }
  else        { src[e] = e - EE; dst[e] = e - EE; }   // self loops
}
__global__ void fill_u32(unsigned int* p, unsigned int v, int n) {
  int i = blockIdx.x * blockDim.x + threadIdx.x;
  if (i < n) p[i] = v;
}
__global__ void fill_f32(float* p, float v, int n) {
  int i = blockIdx.x * blockDim.x + threadIdx.x;
  if (i < n) p[i] = v;
}
__global__ void init_agg(float* __restrict__ agg, const float* __restrict__ bias, int n) {
  int i = blockIdx.x * blockDim.x + threadIdx.x;
  if (i < n) agg[i] = bias[i & (EMB - 1)];
}
__global__ void relu_ip(float* p, int n) {
  int i = blockIdx.x * blockDim.x + threadIdx.x;
  if (i < n) p[i] = fmaxf(p[i], 0.f);
}

// ---------------- GATv2 edge pipeline ----------------
// gid = e*256 + s*4 + h   (S*H == 256)
__global__ void edge_logits(const float* __restrict__ hf, const int* __restrict__ src,
                            const int* __restrict__ dst, const float* __restrict__ att,
                            float* __restrict__ lbuf, unsigned int* __restrict__ mraw)
{
  int gid = blockIdx.x * blockDim.x + threadIdx.x;
  if (gid >= NED * SS * HH) return;
  int h = gid & 3, s = (gid >> 2) & 63, e = gid >> 8;
  int sn = src[e], dn = dst[e];
  const float* ps = hf + ((size_t)(s * NNODE + sn) * EMB + h * CCH);
  const float* pd = hf + ((size_t)(s * NNODE + dn) * EMB + h * CCH);
  const float* pa = att + h * CCH;
  float acc = 0.f;
#pragma unroll 8
  for (int c = 0; c < CCH; ++c) {
    float v = ps[c] + pd[c];
    v = v > 0.f ? v : NEG_SLOPE * v;            // leaky_relu
    acc += pa[c] * v;
  }
  lbuf[gid] = acc;
  atomicMax(&mraw[((size_t)dn * SS + s) * HH + h], ordf(acc));   // segment max (ordered uint)
}
__global__ void edge_exp(const int* __restrict__ dst,
                         float* __restrict__ lbuf, const unsigned int* __restrict__ mraw,
                         float* __restrict__ denom)
{
  int gid = blockIdx.x * blockDim.x + threadIdx.x;
  if (gid >= NED * SS * HH) return;
  int h = gid & 3, s = (gid >> 2) & 63, e = gid >> 8;
  int dn = dst[e];
  size_t seg = ((size_t)dn * SS + s) * HH + h;
  float ex = expf(lbuf[gid] - unordf(mraw[seg]));
  lbuf[gid] = ex;
  atomicAdd(&denom[seg], ex);
}
__global__ void edge_agg(const float* __restrict__ hf, const int* __restrict__ src,
                         const int* __restrict__ dst, const float* __restrict__ lbuf,
                         const float* __restrict__ denom, float* __restrict__ agg)
{
  int gid = blockIdx.x * blockDim.x + threadIdx.x;
  if (gid >= NED * SS * HH) return;
  int h = gid & 3, s = (gid >> 2) & 63, e = gid >> 8;
  int sn = src[e], dn = dst[e];
  float alpha = lbuf[gid] / denom[((size_t)dn * SS + s) * HH + h];
  const float* ps = hf + ((size_t)(s * NNODE + sn) * EMB + h * CCH);
  float*       po = agg + ((size_t)(s * NNODE + dn) * EMB + h * CCH);
#pragma unroll 8
  for (int c = 0; c < CCH; ++c) atomicAdd(&po[c], alpha * ps[c]);
}

// ---------------- LSTM ----------------
__global__ void bias_sum(const float* a, const float* b, float* o) {
  int i = blockIdx.x * blockDim.x + threadIdx.x;
  if (i < G4) o[i] = a[i] + b[i];
}
__global__ void init_hc(const float* h0, const float* c0, float* h, float* c) {
  int i = blockIdx.x * blockDim.x + threadIdx.x;
  if (i < HC) { h[i] = h0[i]; c[i] = c0[i]; }
}
__global__ void lstm_gates(const float* __restrict__ G, float* __restrict__ h,
                           float* __restrict__ c, float* __restrict__ ys)
{
  int i = blockIdx.x * blockDim.x + threadIdx.x;
  if (i >= HC) return;
  int n = i >> 7, j = i & 127;
  const float* g = G + (size_t)n * G4;
  float ig = sigf(g[j]);
  float fg = sigf(g[EMB + j]);
  float gg = tanhf(g[2 * EMB + j]);
  float og = sigf(g[3 * EMB + j]);
  float cn = fg * c[i] + ig * gg;
  float hn = og * tanhf(cn);
  c[i] = cn; h[i] = hn; ys[i] = hn;
}
__global__ void write_hc(const float* h, const float* c, float* out) {
  int i = blockIdx.x * blockDim.x + threadIdx.x;
  if (i < HC) { out[SN + i] = h[i]; out[SN + HC + i] = c[i]; }
}

// ---------------- head ----------------
__global__ void meanpool(const float* __restrict__ ys, float* __restrict__ mp) {
  int i = blockIdx.x * blockDim.x + threadIdx.x;
  if (i >= SS * EMB) return;
  int t = i >> 7, j = i & 127;
  const float* p = ys + (size_t)t * NNODE * EMB + j;
  float s = 0.f;
  for (int n = 0; n < NNODE; ++n) s += p[(size_t)n * EMB];
  mp[i] = s * (1.0f / NNODE);
}
// logits[t][n] = b5 + w5[0:128].relu(gs[t]) + w5[128:256].relu(la[t][n]); softmax over n
__global__ void head_softmax(const float* __restrict__ gs, const float* __restrict__ la,
                             const float* __restrict__ w5, const float* __restrict__ b5,
                             const unsigned char* __restrict__ reach, float* __restrict__ prob)
{
  int t = blockIdx.x, tid = threadIdx.x;
  __shared__ float red[256];
  float lg[2]; lg[0] = lg[1] = -INFINITY;
  float locmax = -INFINITY;
  for (int idx = 0; idx < 2; ++idx) {
    int n = tid + idx * 256;
    if (n < NNODE) {
      float acc = b5[0];
      for (int j = 0; j < EMB; ++j) acc += w5[j]       * fmaxf(gs[(size_t)t * EMB + j], 0.f);
      for (int j = 0; j < EMB; ++j) acc += w5[EMB + j] * fmaxf(la[((size_t)t * NNODE + n) * EMB + j], 0.f);
      if (!reach[(size_t)t * NNODE + n]) acc = -INFINITY;
      lg[idx] = acc;
      locmax = fmaxf(locmax, acc);
    }
  }
  red[tid] = locmax; __syncthreads();
  for (int s = 128; s > 0; s >>= 1) { if (tid < s) red[tid] = fmaxf(red[tid], red[tid + s]); __syncthreads(); }
  float m = red[0]; __syncthreads();
  float ex[2]; float lsum = 0.f;
  for (int idx = 0; idx < 2; ++idx) {
    int n = tid + idx * 256; ex[idx] = 0.f;
    if (n < NNODE) { ex[idx] = expf(lg[idx] - m); lsum += ex[idx]; }
  }
  red[tid] = lsum; __syncthreads();
  for (int s = 128; s > 0; s >>= 1) { if (tid < s) red[tid] += red[tid + s]; __syncthreads(); }
  float dsum = red[0];
  for (int idx = 0; idx < 2; ++idx) {
    int n = tid + idx * 256;
    if (n < NNODE) prob[(size_t)t * NNODE + n] = ex[idx] / dsum;
  }
}

// ---------------- host ----------------
static inline void launch_gemm(const float* A, const float* Bt, int K, int lda, int ldb,
                               const float* A2, const float* Bt2, int K2, int lda2, int ldb2,
                               const float* bias, float* D, int ldd, int M, int N, int relu,
                               hipStream_t st)
{
  int grps = (M / 16) * (N / 64);                 // one wave per 16x64 strip
  int blocks = (grps * 32 + 255) / 256;
  gemm_nt_wmma<<<blocks, 256, 0, st>>>(A, Bt, K, lda, ldb, A2, Bt2, K2, lda2, ldb2,
                                       bias, D, ldd, M, N, relu);
}

extern "C" void kernel_launch(void* const* d_in, const int* in_sizes, int n_in,
                              void* d_out, int out_size, void* d_ws, size_t ws_size,
                              hipStream_t stream)
{
  (void)in_sizes; (void)n_in; (void)out_size; (void)ws_size;
  const float* nfm  = (const float*)d_in[0];
  const int*   ei   = (const int*)d_in[1];
  const unsigned char* reach = (const unsigned char*)d_in[2];
  const float* h0   = (const float*)d_in[3];
  const float* c0   = (const float*)d_in[4];
  const float* gW[3], *gb[3], *gatt[3], *gbias[3];
  for (int i = 0; i < 3; ++i) {
    gW[i]    = (const float*)d_in[5 + 4 * i];
    gb[i]    = (const float*)d_in[6 + 4 * i];
    gatt[i]  = (const float*)d_in[7 + 4 * i];
    gbias[i] = (const float*)d_in[8 + 4 * i];
  }
  const float* Wih = (const float*)d_in[17];
  const float* Whh = (const float*)d_in[18];
  const float* bih = (const float*)d_in[19];
  const float* bhh = (const float*)d_in[20];
  const float* th6_W = (const float*)d_in[21];
  const float* th6_b = (const float*)d_in[22];
  const float* th7_W = (const float*)d_in[23];
  const float* th7_b = (const float*)d_in[24];
  const float* th5_W = (const float*)d_in[25];
  const float* th5_b = (const float*)d_in[26];
  float* out = (float*)d_out;

  // workspace layout
  char* wp = (char*)d_ws;
  auto carve = [&](size_t bytes) { void* r = (void*)wp; wp += (bytes + 255) & ~(size_t)255; return r; };
  int*   srci  = (int*)carve(NED * sizeof(int));
  int*   dsti  = (int*)carve(NED * sizeof(int));
  float* actA  = (float*)carve((size_t)ACT * 4);
  float* actB  = (float*)carve((size_t)ACT * 4);
  float* actH  = (float*)carve((size_t)ACT * 4);
  float* lbuf  = (float*)carve((size_t)NED * SS * HH * 4);
  unsigned int* mraw = (unsigned int*)carve((size_t)NNODE * SS * HH * 4);
  float* denom = (float*)carve((size_t)NNODE * SS * HH * 4);
  float* bsum  = (float*)carve(G4 * 4);
  float* hbuf  = (float*)carve((size_t)HC * 4);
  float* cbuf  = (float*)carve((size_t)HC * 4);
  float* Gbuf  = (float*)carve((size_t)NNODE * G4 * 4);
  float* ysbuf = (float*)carve((size_t)ACT * 4);
  float* mpbuf = (float*)carve((size_t)SS * EMB * 4);
  float* gsbuf = (float*)carve((size_t)SS * EMB * 4);
  float* labuf = (float*)carve((size_t)ACT * 4);

  const int NEG_TOT = NED * SS * HH;   // 2150400
  const int SEG_TOT = NNODE * SS * HH; // 102400

  build_edges<<<(NED + 255) / 256, 256, 0, stream>>>(ei, srci, dsti);

  // ---- 3 GATv2 layers ----
  const float* xin[3]  = { nfm, actA, actB };
  float*       xout[3] = { actA, actB, actA };
  const int    fin[3]  = { 8, EMB, EMB };
  for (int L = 0; L < 3; ++L) {
    // h = x @ W^T + b   (WMMA)
    launch_gemm(xin[L], gW[L], fin[L], fin[L], fin[L],
                nullptr, nullptr, 0, 0, 0, gb[L], actH, EMB, SN, EMB, 0, stream);
    init_agg<<<(ACT + 255) / 256, 256, 0, stream>>>(xout[L], gbias[L], ACT);
    fill_u32<<<(SEG_TOT + 255) / 256, 256, 0, stream>>>(mraw, 0u, SEG_TOT);
    fill_f32<<<(SEG_TOT + 255) / 256, 256, 0, stream>>>(denom, 0.f, SEG_TOT);
    edge_logits<<<(NEG_TOT + 255) / 256, 256, 0, stream>>>(actH, srci, dsti, gatt[L], lbuf, mraw);
    edge_exp<<<(NEG_TOT + 255) / 256, 256, 0, stream>>>(dsti, lbuf, mraw, denom);
    edge_agg<<<(NEG_TOT + 255) / 256, 256, 0, stream>>>(actH, srci, dsti, lbuf, denom, xout[L]);
    if (L < 2) relu_ip<<<(ACT + 255) / 256, 256, 0, stream>>>(xout[L], ACT);
  }
  const float* mu = actA;   // GAT output, LSTM input [T=64, B=400, 128]

  // ---- LSTM (64 sequential steps; fused dual-accumulate WMMA per step) ----
  bias_sum<<<2, 256, 0, stream>>>(bih, bhh, bsum);
  init_hc<<<(HC + 255) / 256, 256, 0, stream>>>(h0, c0, hbuf, cbuf);
  for (int t = 0; t < SS; ++t) {
    launch_gemm(mu + (size_t)t * HC, Wih, EMB, EMB, EMB,
                hbuf, Whh, EMB, EMB, EMB,
                bsum, Gbuf, G4, NNODE, G4, 0, stream);
    lstm_gates<<<(HC + 255) / 256, 256, 0, stream>>>(Gbuf, hbuf, cbuf, ysbuf + (size_t)t * HC);
  }
  write_hc<<<(HC + 255) / 256, 256, 0, stream>>>(hbuf, cbuf, out);

  // ---- head ----
  meanpool<<<(SS * EMB + 255) / 256, 256, 0, stream>>>(ysbuf, mpbuf);
  launch_gemm(mpbuf, th6_W, EMB, EMB, EMB, nullptr, nullptr, 0, 0, 0,
              th6_b, gsbuf, EMB, SS, EMB, 0, stream);
  launch_gemm(ysbuf, th7_W, EMB, EMB, EMB, nullptr, nullptr, 0, 0, 0,
              th7_b, labuf, EMB, SN, EMB, 0, stream);
  head_softmax<<<SS, 256, 0, stream>>>(gsbuf, labuf, th5_W, th5_b, reach, out);
}